// SMPL_11244224381360
// MI455X (gfx1250) — compile-verified
//
#include <hip/hip_runtime.h>
#include <math.h>

// ---------------- problem constants ----------------
#define B_    512
#define V_    6890
#define J_    24
#define NB_   10
#define P_    207            // (J-1)*9
#define KEXT  220            // 207 pose + 10 betas + 1 ones, padded to mult of 4
#define KPAIR (KEXT/2)       // 110 k-pairs (pair-interleaved storage)
#define MPAD  512            // batch dim (multiple of 32)
#define NV3   20670          // V*3
#define VTILES 431           // ceil(6890/16)
#define NPAD  (VTILES*48)    // 20688 padded columns of the fused GEMM

typedef __attribute__((ext_vector_type(2))) float v2f;
typedef __attribute__((ext_vector_type(8))) float v8f;

__constant__ int PARENTS_C[J_] = {-1,0,0,0,1,2,3,4,5,6,7,8,9,9,9,12,13,14,16,17,18,19,20,21};

// ---------------------------------------------------------------------------
// Kernel 0: build augmented B matrix, k-pair interleaved: B2[k/2][n][2]
//   logical rows 0..206   : posedirs[k][n]
//   logical rows 207..216 : shapedirs[(v,c),l]  (col n = v*3+c, l = k-207)
//   logical row  217      : v_template flat
//   rows 218..219 and padding cols : 0
// ---------------------------------------------------------------------------
__global__ void k_bext(const float* __restrict__ posedirs,
                       const float* __restrict__ sdirs,
                       const float* __restrict__ vt,
                       float* __restrict__ Bext2) {
    int idx = blockIdx.x * blockDim.x + threadIdx.x;
    const int total = KPAIR * NPAD * 2;
    if (idx >= total) return;
    int pair = idx / (NPAD * 2);
    int rem  = idx - pair * (NPAD * 2);
    int n    = rem >> 1;
    int k    = pair * 2 + (rem & 1);
    float val = 0.0f;
    if (n < NV3) {
        if (k < P_)             val = posedirs[k * NV3 + n];
        else if (k < P_ + NB_)  val = sdirs[n * NB_ + (k - P_)];
        else if (k == P_ + NB_) val = vt[n];
    }
    Bext2[idx] = val;
}

// ---------------------------------------------------------------------------
// Kernel 1: Rodrigues per (b,j); rot mats + pair-interleaved A: A2[k/2][b][2]
//   logical k: 0..206 = pose_feature (R - I), 207..216 = betas, 217 = 1.0,
//   218..219 = 0
// ---------------------------------------------------------------------------
__device__ __forceinline__ void a2_write(float* A2, int k, int b, float v) {
    A2[((k >> 1) * MPAD + b) * 2 + (k & 1)] = v;
}

__global__ void k_rod(const float* __restrict__ go,
                      const float* __restrict__ bp,
                      const float* __restrict__ betas,
                      float* __restrict__ RotWs,
                      float* __restrict__ AextT2) {
    int idx = blockIdx.x * blockDim.x + threadIdx.x;
    if (idx >= B_ * J_) return;
    int b = idx / J_;
    int j = idx - b * J_;
    const float* aa = (j == 0) ? (go + b * 3) : (bp + b * (J_ - 1) * 3 + (j - 1) * 3);
    float x = aa[0], y = aa[1], z = aa[2];
    float ang = sqrtf(x * x + y * y + z * z) + 1e-8f;
    float inv = 1.0f / ang;
    x *= inv; y *= inv; z *= inv;
    float s = sinf(ang);
    float c = cosf(ang);
    float ic = 1.0f - c;
    float R[9];
    R[0] = c + ic * x * x;      R[1] = ic * x * y - s * z;  R[2] = ic * x * z + s * y;
    R[3] = ic * x * y + s * z;  R[4] = c + ic * y * y;      R[5] = ic * y * z - s * x;
    R[6] = ic * x * z - s * y;  R[7] = ic * y * z + s * x;  R[8] = c + ic * z * z;
    #pragma unroll
    for (int e = 0; e < 9; ++e) RotWs[idx * 9 + e] = R[e];
    if (j > 0) {
        int kb = (j - 1) * 9;
        #pragma unroll
        for (int e = 0; e < 9; ++e)
            a2_write(AextT2, kb + e, b, R[e] - ((e % 4 == 0) ? 1.0f : 0.0f));
    } else {
        #pragma unroll
        for (int l = 0; l < NB_; ++l)
            a2_write(AextT2, P_ + l, b, betas[b * NB_ + l]);
        a2_write(AextT2, 217, b, 1.0f);
        a2_write(AextT2, 218, b, 0.0f);
        a2_write(AextT2, 219, b, 0.0f);
    }
}

// ---------------------------------------------------------------------------
// Kernel 2: batch-independent joint-regressor reductions
//   Jt[j,c]   = sum_v Jreg[j,v] * v_template[v,c]
//   Js[j,c,l] = sum_v Jreg[j,v] * shapedirs[v,c,l]
// ---------------------------------------------------------------------------
__global__ void k_jreg(const float* __restrict__ Jreg,
                       const float* __restrict__ vt,
                       const float* __restrict__ sdirs,
                       float* __restrict__ Jt,
                       float* __restrict__ Js) {
    int jc = blockIdx.x;           // 0..71
    int j = jc / 3, c = jc - j * 3;
    float acc[NB_ + 1] = {};
    for (int v = threadIdx.x; v < V_; v += 256) {
        float r = Jreg[j * V_ + v];
        acc[0] += r * vt[v * 3 + c];
        const float* sd = &sdirs[(v * 3 + c) * NB_];
        #pragma unroll
        for (int l = 0; l < NB_; ++l) acc[1 + l] += r * sd[l];
    }
    __shared__ float red[256];
    for (int i = 0; i < NB_ + 1; ++i) {
        red[threadIdx.x] = acc[i];
        __syncthreads();
        for (int s = 128; s > 0; s >>= 1) {
            if (threadIdx.x < s) red[threadIdx.x] += red[threadIdx.x + s];
            __syncthreads();
        }
        if (threadIdx.x == 0) {
            if (i == 0) Jt[j * 3 + c] = red[0];
            else        Js[(j * 3 + c) * NB_ + (i - 1)] = red[0];
        }
        __syncthreads();
    }
}

// ---------------------------------------------------------------------------
// Kernel 3: per-batch joints + kinematic chain -> A_rel [b][24][12], joints_out
// ---------------------------------------------------------------------------
__global__ void k_chain(const float* __restrict__ betas,
                        const float* __restrict__ transl,
                        const float* __restrict__ RotWs,
                        const float* __restrict__ Jt,
                        const float* __restrict__ Js,
                        float* __restrict__ Arel,
                        float* __restrict__ jout) {
    int b = blockIdx.x * blockDim.x + threadIdx.x;
    if (b >= B_) return;
    float jnt[J_][3];
    for (int j = 0; j < J_; ++j)
        for (int c = 0; c < 3; ++c) {
            float s = Jt[j * 3 + c];
            #pragma unroll
            for (int l = 0; l < NB_; ++l)
                s += betas[b * NB_ + l] * Js[(j * 3 + c) * NB_ + l];
            jnt[j][c] = s;
        }
    float AR[J_][9];
    float At[J_][3];
    for (int i = 0; i < J_; ++i) {
        const float* R = &RotWs[(b * J_ + i) * 9];
        int p = PARENTS_C[i];
        float rel[3];
        if (i == 0) { rel[0] = jnt[0][0]; rel[1] = jnt[0][1]; rel[2] = jnt[0][2]; }
        else { for (int c = 0; c < 3; ++c) rel[c] = jnt[i][c] - jnt[p][c]; }
        if (i == 0) {
            for (int e = 0; e < 9; ++e) AR[0][e] = R[e];
            for (int c = 0; c < 3; ++c) At[0][c] = rel[c];
        } else {
            for (int m = 0; m < 3; ++m) {
                for (int n = 0; n < 3; ++n) {
                    float s = 0.0f;
                    for (int k = 0; k < 3; ++k) s += AR[p][m * 3 + k] * R[k * 3 + n];
                    AR[i][m * 3 + n] = s;
                }
                float t = At[p][m];
                for (int k = 0; k < 3; ++k) t += AR[p][m * 3 + k] * rel[k];
                At[i][m] = t;
            }
        }
    }
    for (int i = 0; i < J_; ++i) {
        for (int m = 0; m < 3; ++m) {
            float tr = At[i][m];
            for (int k = 0; k < 3; ++k) tr -= AR[i][m * 3 + k] * jnt[i][k];
            float* ar = &Arel[b * (J_ * 12) + i * 12 + m * 4];
            ar[0] = AR[i][m * 3 + 0];
            ar[1] = AR[i][m * 3 + 1];
            ar[2] = AR[i][m * 3 + 2];
            ar[3] = tr;
            jout[(b * J_ + i) * 3 + m] = At[i][m] + transl[b * 3 + m];
        }
    }
}

// ---------------------------------------------------------------------------
// Kernel 4: fused GEMM (fp32 WMMA 16x16x4) + LBS skinning epilogue
//   Block = 3 waves (96 thr). Tile: 32 batches x 16 vertices.
//   Each wave owns one 16-col n-tile and TWO m-tiles (dual accumulators):
//   every B fragment (one global_load_b64) feeds two WMMAs; A fragments are
//   single b64 loads from the pair-interleaved layout.
// ---------------------------------------------------------------------------
__global__ __launch_bounds__(96) void k_gemm_skin(
    const float* __restrict__ AextT2,  // [KPAIR][MPAD][2]
    const float* __restrict__ Bext2,   // [KPAIR][NPAD][2]
    const float* __restrict__ Arel,    // [B][24][12]
    const float* __restrict__ lbs,     // [V][24]
    const float* __restrict__ transl,  // [B][3]
    float* __restrict__ vout) {        // [B][V][3]
    const int vtile = blockIdx.x;
    const int m0 = blockIdx.y * 32;
    const int tid = threadIdx.x;
    const int wave = tid >> 5;
    const int lane = tid & 31;
    const int half = lane >> 4;
    const int l16 = lane & 15;
    const int n0 = vtile * 48 + wave * 16;

    const v2f* __restrict__ A2 = (const v2f*)AextT2;
    const v2f* __restrict__ B2 = (const v2f*)Bext2;

    v8f c0 = {0.f, 0.f, 0.f, 0.f, 0.f, 0.f, 0.f, 0.f};
    v8f c1 = {0.f, 0.f, 0.f, 0.f, 0.f, 0.f, 0.f, 0.f};
    const int rowA0 = m0 + l16;
    const int rowA1 = m0 + 16 + l16;
    const int colB  = n0 + l16;
    for (int k0 = 0; k0 < KEXT; k0 += 4) {
        int pair = (k0 >> 1) + half;           // k = 2*pair, 2*pair+1
        v2f a0 = A2[pair * MPAD + rowA0];      // global_load_b64
        v2f a1 = A2[pair * MPAD + rowA1];      // global_load_b64
        v2f bb = B2[pair * NPAD + colB];       // global_load_b64 (shared by both)
        c0 = __builtin_amdgcn_wmma_f32_16x16x4_f32(false, a0, false, bb,
                                                   (short)0, c0, false, false);
        c1 = __builtin_amdgcn_wmma_f32_16x16x4_f32(false, a1, false, bb,
                                                   (short)0, c1, false, false);
    }

    __shared__ float ctile[32 * 48];     // v_posed tile [bi][local col]
    __shared__ float arel_s[32 * 288];   // A_rel for 32 batches
    __shared__ float w_s[16 * 24];       // lbs weights for 16 vertices
    #pragma unroll
    for (int r = 0; r < 8; ++r) {
        ctile[(r + 8 * half) * 48 + wave * 16 + l16]        = c0[r];
        ctile[(16 + r + 8 * half) * 48 + wave * 16 + l16]   = c1[r];
    }
    for (int i = tid; i < 32 * 288; i += 96)
        arel_s[i] = Arel[(m0 + (i / 288)) * 288 + (i % 288)];
    for (int i = tid; i < 16 * 24; i += 96) {
        int vi = i / 24;
        int v = vtile * 16 + vi;
        w_s[i] = (v < V_) ? lbs[v * J_ + (i % 24)] : 0.0f;
    }
    __syncthreads();

    for (int p = tid; p < 512; p += 96) {
        int bi = p >> 4, vi = p & 15;
        int v = vtile * 16 + vi;
        if (v >= V_) continue;
        int b = m0 + bi;
        float T[12];
        #pragma unroll
        for (int e = 0; e < 12; ++e) T[e] = 0.0f;
        for (int j = 0; j < J_; ++j) {
            float wj = w_s[vi * 24 + j];
            const float* ar = &arel_s[bi * 288 + j * 12];
            #pragma unroll
            for (int e = 0; e < 12; ++e) T[e] += wj * ar[e];
        }
        float x = ctile[bi * 48 + vi * 3 + 0];
        float y = ctile[bi * 48 + vi * 3 + 1];
        float z = ctile[bi * 48 + vi * 3 + 2];
        #pragma unroll
        for (int m = 0; m < 3; ++m) {
            float o = T[m * 4 + 0] * x + T[m * 4 + 1] * y + T[m * 4 + 2] * z
                    + T[m * 4 + 3] + transl[b * 3 + m];
            vout[((size_t)b * V_ + v) * 3 + m] = o;
        }
    }
}

// ---------------------------------------------------------------------------
extern "C" void kernel_launch(void* const* d_in, const int* in_sizes, int n_in,
                              void* d_out, int out_size, void* d_ws, size_t ws_size,
                              hipStream_t stream) {
    const float* betas     = (const float*)d_in[0];
    const float* body_pose = (const float*)d_in[1];
    const float* go        = (const float*)d_in[2];
    const float* transl    = (const float*)d_in[3];
    const float* vt        = (const float*)d_in[4];
    const float* sdirs     = (const float*)d_in[5];
    const float* posedirs  = (const float*)d_in[6];
    const float* Jreg      = (const float*)d_in[7];
    const float* lbs       = (const float*)d_in[8];

    float* out  = (float*)d_out;
    float* vout = out;                              // B*V*3 vertices
    float* jout = out + (size_t)B_ * V_ * 3;        // B*J*3 joints

    // workspace layout (floats): ~19.7 MB total
    float* ws     = (float*)d_ws;
    float* Bext2  = ws;                             // KEXT*NPAD   = 4,551,360
    float* AextT2 = Bext2  + (size_t)KEXT * NPAD;   // KEXT*MPAD   =   112,640
    float* RotWs  = AextT2 + (size_t)KEXT * MPAD;   // B*J*9       =   110,592
    float* Arel   = RotWs  + (size_t)B_ * J_ * 9;   // B*J*12      =   147,456
    float* Jt     = Arel   + (size_t)B_ * J_ * 12;  // 72
    float* Js     = Jt + 72;                        // 720

    k_bext<<<(KEXT * NPAD + 255) / 256, 256, 0, stream>>>(posedirs, sdirs, vt, Bext2);
    k_rod<<<(B_ * J_ + 255) / 256, 256, 0, stream>>>(go, body_pose, betas, RotWs, AextT2);
    k_jreg<<<J_ * 3, 256, 0, stream>>>(Jreg, vt, sdirs, Jt, Js);
    k_chain<<<(B_ + 63) / 64, 64, 0, stream>>>(betas, transl, RotWs, Jt, Js, Arel, jout);
    k_gemm_skin<<<dim3(VTILES, B_ / 32), 96, 0, stream>>>(AextT2, Bext2, Arel, lbs, transl, vout);
}